// ODEBlock_28389733826565
// MI455X (gfx1250) — compile-verified
//
#include <hip/hip_runtime.h>
#include <math.h>

// CDNA5 / gfx1250: fp32 WMMA Neural-ODE (DOPRI5, 100 fixed steps).
// Each workgroup owns 64 rows of y; weights live in LDS (column-major for
// single-ds_load_b64 WMMA-B tiles); per-wave 8KB LDS staging converts
// C-layout register slabs into WMMA-A operands. y,k1..k6 stay in VGPRs.

typedef __attribute__((ext_vector_type(2))) float v2f;
typedef __attribute__((ext_vector_type(8))) float v8f;

#define D        128
#define WAVES    4
#define THREADS  128
#define ROWS_WG  64
#define NJ       8     // 8 N-tiles of 16 cols
#define NKC      32    // 32 K-chunks of 4
#define NSTEPS   100
#define DTF      0.01f

#if __has_builtin(__builtin_amdgcn_tanhf)
#define TANHF(x) __builtin_amdgcn_tanhf(x)
#elif __has_builtin(__builtin_amdgcn_tanh_f32)
#define TANHF(x) __builtin_amdgcn_tanh_f32(x)
#else
#define TANHF(x) tanhf(x)
#endif

// GEMM: acc[j] (16x16 f32 C-layout tiles) += A(16x128 row-major LDS) * B(128x128 col-major LDS)
__device__ __forceinline__ void wmma_gemm(const float* __restrict__ sA,
                                          const float* __restrict__ sB,
                                          v8f* acc, int lm, int hi2) {
  for (int kc = 0; kc < NKC; ++kc) {
    // A tile 16x4: lane l -> row lm, K = 4kc + 2*(l/16) (+1 in second dword)
    v2f a = *(const v2f*)(sA + lm * D + (kc << 2) + hi2);
#pragma unroll
    for (int j = 0; j < NJ; ++j) {
      // B tile 4x16 from col-major: lane l -> col (16j+lm), K = 4kc + 2*(l/16) (+1)
      v2f b = *(const v2f*)(sB + (((j << 4) + lm) << 7) + (kc << 2) + hi2);
      acc[j] = __builtin_amdgcn_wmma_f32_16x16x4_f32(
          /*neg_a=*/false, a, /*neg_b=*/false, b,
          /*c_mod=*/(short)0, acc[j], /*reuse_a=*/false, /*reuse_b=*/false);
    }
  }
}

// Store a C-layout register slab (8 tiles of 16x16) to row-major 16x128 LDS.
__device__ __forceinline__ void store_slab(float* __restrict__ sS,
                                           const v8f* z, int lm, int hi) {
#pragma unroll
  for (int j = 0; j < NJ; ++j)
#pragma unroll
    for (int r = 0; r < 8; ++r)
      sS[(r + (hi << 3)) * D + (j << 4) + lm] = z[j][r];
}

// kout = tanh(slab_in_sS @ W1 + b1) @ W2 + b2  (slab must already be in sS)
__device__ __forceinline__ void f_eval(float* __restrict__ sS,
                                       const float* __restrict__ sW1,
                                       const float* __restrict__ sW2,
                                       const float* __restrict__ b1v,
                                       const float* __restrict__ b2v,
                                       v8f* kout, int lm, int hi, int hi2) {
  v8f h[NJ];
#pragma unroll
  for (int j = 0; j < NJ; ++j)
#pragma unroll
    for (int r = 0; r < 8; ++r) h[j][r] = b1v[j];
  wmma_gemm(sS, sW1, h, lm, hi2);
#pragma unroll
  for (int j = 0; j < NJ; ++j)
#pragma unroll
    for (int r = 0; r < 8; ++r) h[j][r] = TANHF(h[j][r]);
  store_slab(sS, h, lm, hi);   // staging buffer reused: z fully consumed above
#pragma unroll
  for (int j = 0; j < NJ; ++j)
#pragma unroll
    for (int r = 0; r < 8; ++r) kout[j][r] = b2v[j];
  wmma_gemm(sS, sW2, kout, lm, hi2);
}

__global__ __launch_bounds__(THREADS, 1)
void ode_dopri5_kernel(const float* __restrict__ xg,
                       const float* __restrict__ W1g,
                       const float* __restrict__ b1g,
                       const float* __restrict__ W2g,
                       const float* __restrict__ b2g,
                       float* __restrict__ outg) {
  extern __shared__ float smem[];
  float* sW1 = smem;               // 16384 floats, col-major W1
  float* sW2 = smem + D * D;       // 16384 floats, col-major W2
  float* sStageAll = smem + 2 * D * D;  // WAVES * 2048 floats

  const int tid  = threadIdx.x;
  const int wave = tid >> 5;
  const int lane = tid & 31;
  const int lm   = lane & 15;
  const int hi   = lane >> 4;
  const int hi2  = hi << 1;
  float* sS = sStageAll + wave * (16 * D);

  // Cooperative transpose-load of weights into LDS (column-major).
  for (int i = tid; i < D * D; i += THREADS) {
    int k = i >> 7, n = i & (D - 1);
    sW1[(n << 7) + k] = W1g[i];
    sW2[(n << 7) + k] = W2g[i];
  }
  __syncthreads();

  float b1v[NJ], b2v[NJ];
#pragma unroll
  for (int j = 0; j < NJ; ++j) {
    b1v[j] = b1g[(j << 4) + lm];
    b2v[j] = b2g[(j << 4) + lm];
  }

  const int rowBase = blockIdx.x * ROWS_WG + wave * 16;

  // Load y slab (16x128) into C-layout register tiles.
  v8f y[NJ];
#pragma unroll
  for (int j = 0; j < NJ; ++j)
#pragma unroll
    for (int r = 0; r < 8; ++r)
      y[j][r] = xg[(rowBase + r + (hi << 3)) * D + (j << 4) + lm];

  v8f k1[NJ], k2[NJ], k3[NJ], k4[NJ], k5[NJ], k6[NJ], z[NJ];

  for (int step = 0; step < NSTEPS; ++step) {
    // stage 1: k1 = f(y)
    store_slab(sS, y, lm, hi);
    f_eval(sS, sW1, sW2, b1v, b2v, k1, lm, hi, hi2);

    // stage 2
#pragma unroll
    for (int j = 0; j < NJ; ++j)
#pragma unroll
      for (int r = 0; r < 8; ++r)
        z[j][r] = y[j][r] + DTF * (0.2f * k1[j][r]);
    store_slab(sS, z, lm, hi);
    f_eval(sS, sW1, sW2, b1v, b2v, k2, lm, hi, hi2);

    // stage 3
#pragma unroll
    for (int j = 0; j < NJ; ++j)
#pragma unroll
      for (int r = 0; r < 8; ++r)
        z[j][r] = y[j][r] + DTF * (0.075f * k1[j][r] + 0.225f * k2[j][r]);
    store_slab(sS, z, lm, hi);
    f_eval(sS, sW1, sW2, b1v, b2v, k3, lm, hi, hi2);

    // stage 4
#pragma unroll
    for (int j = 0; j < NJ; ++j)
#pragma unroll
      for (int r = 0; r < 8; ++r)
        z[j][r] = y[j][r] + DTF * ((44.0f / 45.0f) * k1[j][r]
                                 - (56.0f / 15.0f) * k2[j][r]
                                 + (32.0f / 9.0f)  * k3[j][r]);
    store_slab(sS, z, lm, hi);
    f_eval(sS, sW1, sW2, b1v, b2v, k4, lm, hi, hi2);

    // stage 5
#pragma unroll
    for (int j = 0; j < NJ; ++j)
#pragma unroll
      for (int r = 0; r < 8; ++r)
        z[j][r] = y[j][r] + DTF * ((19372.0f / 6561.0f) * k1[j][r]
                                 - (25360.0f / 2187.0f) * k2[j][r]
                                 + (64448.0f / 6561.0f) * k3[j][r]
                                 - (212.0f / 729.0f)    * k4[j][r]);
    store_slab(sS, z, lm, hi);
    f_eval(sS, sW1, sW2, b1v, b2v, k5, lm, hi, hi2);

    // stage 6
#pragma unroll
    for (int j = 0; j < NJ; ++j)
#pragma unroll
      for (int r = 0; r < 8; ++r)
        z[j][r] = y[j][r] + DTF * ((9017.0f / 3168.0f)  * k1[j][r]
                                 - (355.0f / 33.0f)     * k2[j][r]
                                 + (46732.0f / 5247.0f) * k3[j][r]
                                 + (49.0f / 176.0f)     * k4[j][r]
                                 - (5103.0f / 18656.0f) * k5[j][r]);
    store_slab(sS, z, lm, hi);
    f_eval(sS, sW1, sW2, b1v, b2v, k6, lm, hi, hi2);

    // 5th-order combine
#pragma unroll
    for (int j = 0; j < NJ; ++j)
#pragma unroll
      for (int r = 0; r < 8; ++r)
        y[j][r] += DTF * ((35.0f / 384.0f)    * k1[j][r]
                        + (500.0f / 1113.0f)  * k3[j][r]
                        + (125.0f / 192.0f)   * k4[j][r]
                        - (2187.0f / 6784.0f) * k5[j][r]
                        + (11.0f / 84.0f)     * k6[j][r]);
  }

#pragma unroll
  for (int j = 0; j < NJ; ++j)
#pragma unroll
    for (int r = 0; r < 8; ++r)
      outg[(rowBase + r + (hi << 3)) * D + (j << 4) + lm] = y[j][r];
}

extern "C" void kernel_launch(void* const* d_in, const int* in_sizes, int n_in,
                              void* d_out, int out_size, void* d_ws, size_t ws_size,
                              hipStream_t stream) {
  const float* x  = (const float*)d_in[0];
  const float* W1 = (const float*)d_in[1];
  const float* b1 = (const float*)d_in[2];
  const float* W2 = (const float*)d_in[3];
  const float* b2 = (const float*)d_in[4];
  float* out = (float*)d_out;

  const int batch = in_sizes[0] / D;           // 32768
  const int grid  = batch / ROWS_WG;           // 512 workgroups
  const size_t shmem = (size_t)(2 * D * D + WAVES * 16 * D) * sizeof(float); // 160 KB

  ode_dopri5_kernel<<<dim3(grid), dim3(THREADS), shmem, stream>>>(x, W1, b1, W2, b2, out);
}